// SVAE_LDS_86861418594793
// MI455X (gfx1250) — compile-verified
//
#include <hip/hip_runtime.h>
#include <stdint.h>

#define LDIM 4
#define OBS  10
#define BS   1024
#define TT   1024
#define NPOS ((size_t)BS * (size_t)TT)
#define EPS_SAMPLE 1e-4f

typedef __attribute__((ext_vector_type(2))) float v2f;
typedef __attribute__((ext_vector_type(4))) float v4f;
typedef __attribute__((ext_vector_type(8))) float v8f;

// D(16x16 f32) = A(16x4 f32) * B(4x16 f32) + C   -- V_WMMA_F32_16X16X4_F32
__device__ __forceinline__ v8f wmma4(v2f a, v2f b, v8f c) {
  return __builtin_amdgcn_wmma_f32_16x16x4_f32(false, a, false, b, (short)0, c,
                                               false, false);
}

__device__ __forceinline__ void st_nt4(float* p, v4f v) {
  __builtin_nontemporal_store(v, (v4f*)p);
}

__device__ __forceinline__ uint32_t hash_u32(uint32_t x) {
  x ^= x >> 16; x *= 0x7feb352du;
  x ^= x >> 15; x *= 0x846ca68bu;
  x ^= x >> 16;
  return x;
}

// ---------------------------------------------------------------------------
// Prep: per-step shared noise u[t][4] (Box-Muller) and Q = L L^T from kf_Q
// (tfp fill_triangular packing:
//  L = [[e^q4,0,0,0],[q8,e^q9,0,0],[q7,q6,e^q5,0],[q3,q2,q1,e^q0]])
// ---------------------------------------------------------------------------
__global__ void svae_prep(const float* __restrict__ kq,
                          const int* __restrict__ seed,
                          float* __restrict__ u_ws, float* __restrict__ Q_ws) {
  int t = blockIdx.x * blockDim.x + threadIdx.x;
  if (t < TT) {
    uint32_t s = (uint32_t)seed[0] * 0x9e3779b9u + 0xA511E9B3u;
#pragma unroll
    for (int p = 0; p < 2; ++p) {
      uint32_t h1 = hash_u32(((uint32_t)t * 2u + (uint32_t)p) * 2654435761u ^ s);
      uint32_t h2 = hash_u32(h1 ^ 0x63d83595u);
      float u1 = ((float)(h1 >> 8) + 0.5f) * (1.0f / 16777216.0f);
      float u2 = (float)(h2 >> 8) * (1.0f / 16777216.0f);
      float r = sqrtf(-2.0f * __logf(u1));
      float ang = 6.28318530718f * u2;
      u_ws[t * 4 + p * 2 + 0] = r * __cosf(ang);
      u_ws[t * 4 + p * 2 + 1] = r * __sinf(ang);
    }
  }
  if (t == 0) {
    float L[4][4] = {};
    L[0][0] = __expf(kq[4]);
    L[1][0] = kq[8]; L[1][1] = __expf(kq[9]);
    L[2][0] = kq[7]; L[2][1] = kq[6]; L[2][2] = __expf(kq[5]);
    L[3][0] = kq[3]; L[3][1] = kq[2]; L[3][2] = kq[1]; L[3][3] = __expf(kq[0]);
#pragma unroll
    for (int i = 0; i < 4; ++i)
#pragma unroll
      for (int j = 0; j < 4; ++j) {
        float acc = 0.f;
#pragma unroll
        for (int k = 0; k < 4; ++k) acc += L[i][k] * L[j][k];
        Q_ws[i * 4 + j] = acc;
      }
  }
}

// ---------------------------------------------------------------------------
// Encoder: tiles of 16 positions (fixed b, 16 consecutive t).
// Layer1 [10->6] via 3x V_WMMA_F32_16X16X4_F32 (K padded 10->12), relu.
// Fully branchless operand/bias builds: clamped addresses + value masking.
// LDS transpose of C-layout, then scalar [6->4]x2, sigma = exp(0.5*logvar).
// ---------------------------------------------------------------------------
__global__ void __launch_bounds__(256) svae_encoder(
    const float* __restrict__ x,
    const float* __restrict__ w1, const float* __restrict__ b1,
    const float* __restrict__ wx, const float* __restrict__ bx,
    const float* __restrict__ wl, const float* __restrict__ bl,
    float* __restrict__ zm_ws, float* __restrict__ sg_ws) {
  __shared__ float hbuf[8][256];
  const int lane = threadIdx.x & 31;
  const int wave = threadIdx.x >> 5;
  const int tile = blockIdx.x * 8 + wave;   // 65536 tiles
  const int b  = tile >> 6;                 // 64 tiles per batch row
  const int t0 = (tile & 63) << 4;
  const int m  = lane & 15;                 // row (A) / col (B,C) index
  const int hi = lane >> 4;
  const bool mok = (m < 6);
  const int col = mok ? m : 0;              // clamped weight column

  const float* xp = x + ((size_t)b * TT + (size_t)(t0 + m)) * OBS;

  v8f c = {0.f, 0.f, 0.f, 0.f, 0.f, 0.f, 0.f, 0.f};
#pragma unroll
  for (int kc = 0; kc < 3; ++kc) {
    const int k0 = kc * 4 + hi * 2;
    const bool kok = (k0 < OBS);            // pair (k0,k0+1) fully in-bounds
    // A fragment: one aligned b64 load, value-masked
    v2f a = *(const v2f*)(xp + (kok ? k0 : 0));
    a.x = kok ? a.x : 0.f;
    a.y = kok ? a.y : 0.f;
    // B fragment: clamped rows/col, value-masked
    const int r0 = kok ? k0 : 0;
    const bool vld = kok && mok;
    const float w0 = w1[r0 * 6 + col];
    const float w1v = w1[(r0 + 1) * 6 + col];
    v2f w;
    w.x = vld ? w0 : 0.f;
    w.y = vld ? w1v : 0.f;
    c = wmma4(a, w, c);
  }
  const float braw = b1[col];               // clamped, unconditional
  const float bias = mok ? braw : 0.f;
#pragma unroll
  for (int r = 0; r < 8; ++r) {
    float h = c[r] + bias;
    hbuf[wave][(r + 8 * hi) * 16 + m] = h > 0.f ? h : 0.f;
  }
  __syncthreads();

  if (hi == 0) {  // lanes 0..15: one position each
    float zm[4], lv[4];
#pragma unroll
    for (int j = 0; j < 4; ++j) { zm[j] = bx[j]; lv[j] = bl[j]; }
#pragma unroll
    for (int k = 0; k < 6; ++k) {
      const float h = hbuf[wave][m * 16 + k];
#pragma unroll
      for (int j = 0; j < 4; ++j) {
        zm[j] += h * wx[k * 4 + j];
        lv[j] += h * wl[k * 4 + j];
      }
    }
    const size_t o = ((size_t)(t0 + m) * BS + (size_t)b) * 4;
    *(v4f*)(zm_ws + o) = (v4f){zm[0], zm[1], zm[2], zm[3]};
    *(v4f*)(sg_ws + o) = (v4f){__expf(0.5f * lv[0]), __expf(0.5f * lv[1]),
                               __expf(0.5f * lv[2]), __expf(0.5f * lv[3])};
  }
}

// ---------------------------------------------------------------------------
// Kalman scan: one lane per batch chain (1024 lanes), fully register-resident
// 4x4 algebra, T=1024 serial steps. Final outputs streamed with NT hint
// (write-only, never re-read; keep L2 for zm/sg/zh producer->consumer data).
// ---------------------------------------------------------------------------
__global__ void __launch_bounds__(128) svae_scan(
    const float* __restrict__ zm_ws, const float* __restrict__ sg_ws,
    const float* __restrict__ u_ws,  const float* __restrict__ Q_ws,
    const float* __restrict__ kfA,   const float* __restrict__ kfb,
    float* __restrict__ out, float* __restrict__ zh_ws) {
  const int b = blockIdx.x * blockDim.x + threadIdx.x;  // 0..1023

  float A[16], Q[16], bv[4];
#pragma unroll
  for (int i = 0; i < 16; ++i) { A[i] = kfA[i]; Q[i] = Q_ws[i]; }
#pragma unroll
  for (int i = 0; i < 4; ++i) bv[i] = kfb[i];

  float* q_mean = out + 10 * NPOS;
  float* q_cov  = out + 14 * NPOS;
  float* p_mean = out + 30 * NPOS;
  float* p_cov  = out + 34 * NPOS;

  float mc[4] = {0.f, 0.f, 0.f, 0.f};
  float P[16];
#pragma unroll
  for (int i = 0; i < 16; ++i) P[i] = (i % 5 == 0) ? 1.f : 0.f;

  for (int t = 0; t < TT; ++t) {
    const size_t idx = (size_t)t * BS + (size_t)b;
    const v4f zm4 = *(const v4f*)(zm_ws + idx * 4);
    const v4f sg4 = *(const v4f*)(sg_ws + idx * 4);
    const v4f u4  = *(const v4f*)(u_ws + (size_t)t * 4);
    const float zm[4] = {zm4.x, zm4.y, zm4.z, zm4.w};
    const float sg[4] = {sg4.x, sg4.y, sg4.z, sg4.w};
    const float u[4]  = {u4.x, u4.y, u4.z, u4.w};

    // mp = m @ A + b
    float mp[4];
#pragma unroll
    for (int i = 0; i < 4; ++i) {
      float acc = bv[i];
#pragma unroll
      for (int k = 0; k < 4; ++k) acc += mc[k] * A[k * 4 + i];
      mp[i] = acc;
    }
    // Pp = A P A^T + Q
    float T1[16];
#pragma unroll
    for (int k = 0; k < 4; ++k)
#pragma unroll
      for (int j = 0; j < 4; ++j) {
        float acc = 0.f;
#pragma unroll
        for (int l = 0; l < 4; ++l) acc += P[k * 4 + l] * A[j * 4 + l];
        T1[k * 4 + j] = acc;
      }
    float Pp[16];
#pragma unroll
    for (int i = 0; i < 4; ++i)
#pragma unroll
      for (int j = 0; j < 4; ++j) {
        float acc = Q[i * 4 + j];
#pragma unroll
        for (int k = 0; k < 4; ++k) acc += A[i * 4 + k] * T1[k * 4 + j];
        Pp[i * 4 + j] = acc;
      }
    // S = Pp + diag(sigma)
    float S[16];
#pragma unroll
    for (int i = 0; i < 16; ++i) S[i] = Pp[i];
    S[0] += sg[0]; S[5] += sg[1]; S[10] += sg[2]; S[15] += sg[3];

    // 4x4 inverse by cofactors
    const float s0 = S[0] * S[5]  - S[4] * S[1];
    const float s1 = S[0] * S[6]  - S[4] * S[2];
    const float s2 = S[0] * S[7]  - S[4] * S[3];
    const float s3 = S[1] * S[6]  - S[5] * S[2];
    const float s4 = S[1] * S[7]  - S[5] * S[3];
    const float s5 = S[2] * S[7]  - S[6] * S[3];
    const float c5 = S[10] * S[15] - S[14] * S[11];
    const float c4 = S[9]  * S[15] - S[13] * S[11];
    const float c3 = S[9]  * S[14] - S[13] * S[10];
    const float c2 = S[8]  * S[15] - S[12] * S[11];
    const float c1 = S[8]  * S[14] - S[12] * S[10];
    const float c0 = S[8]  * S[13] - S[12] * S[9];
    const float invdet = 1.0f / (s0 * c5 - s1 * c4 + s2 * c3 + s3 * c2 - s4 * c1 + s5 * c0);
    float IV[16];
    IV[0]  = ( S[5]  * c5 - S[6]  * c4 + S[7]  * c3) * invdet;
    IV[1]  = (-S[1]  * c5 + S[2]  * c4 - S[3]  * c3) * invdet;
    IV[2]  = ( S[13] * s5 - S[14] * s4 + S[15] * s3) * invdet;
    IV[3]  = (-S[9]  * s5 + S[10] * s4 - S[11] * s3) * invdet;
    IV[4]  = (-S[4]  * c5 + S[6]  * c2 - S[7]  * c1) * invdet;
    IV[5]  = ( S[0]  * c5 - S[2]  * c2 + S[3]  * c1) * invdet;
    IV[6]  = (-S[12] * s5 + S[14] * s2 - S[15] * s1) * invdet;
    IV[7]  = ( S[8]  * s5 - S[10] * s2 + S[11] * s1) * invdet;
    IV[8]  = ( S[4]  * c4 - S[5]  * c2 + S[7]  * c0) * invdet;
    IV[9]  = (-S[0]  * c4 + S[1]  * c2 - S[3]  * c0) * invdet;
    IV[10] = ( S[12] * s4 - S[13] * s2 + S[15] * s0) * invdet;
    IV[11] = (-S[8]  * s4 + S[9]  * s2 - S[11] * s0) * invdet;
    IV[12] = (-S[4]  * c3 + S[5]  * c1 - S[6]  * c0) * invdet;
    IV[13] = ( S[0]  * c3 - S[1]  * c1 + S[2]  * c0) * invdet;
    IV[14] = (-S[12] * s3 + S[13] * s1 - S[14] * s0) * invdet;
    IV[15] = ( S[8]  * s3 - S[9]  * s1 + S[10] * s0) * invdet;

    // K = Pp @ inv(S)
    float K[16];
#pragma unroll
    for (int i = 0; i < 4; ++i)
#pragma unroll
      for (int j = 0; j < 4; ++j) {
        float acc = 0.f;
#pragma unroll
        for (int k = 0; k < 4; ++k) acc += Pp[i * 4 + k] * IV[k * 4 + j];
        K[i * 4 + j] = acc;
      }
    // mu = mp + K (zm - mp)
    float dz[4];
#pragma unroll
    for (int j = 0; j < 4; ++j) dz[j] = zm[j] - mp[j];
    float mu[4];
#pragma unroll
    for (int i = 0; i < 4; ++i) {
      float acc = mp[i];
#pragma unroll
      for (int j = 0; j < 4; ++j) acc += K[i * 4 + j] * dz[j];
      mu[i] = acc;
    }
    // Pn = (I - K) Pp
    float Pn[16];
#pragma unroll
    for (int i = 0; i < 4; ++i)
#pragma unroll
      for (int j = 0; j < 4; ++j) {
        float acc = Pp[i * 4 + j];
#pragma unroll
        for (int k = 0; k < 4; ++k) acc -= K[i * 4 + k] * Pp[k * 4 + j];
        Pn[i * 4 + j] = acc;
      }

    // outputs (non-temporal: write-only streams)
    st_nt4(q_mean + idx * 4, (v4f){mu[0], mu[1], mu[2], mu[3]});
    st_nt4(p_mean + idx * 4, (v4f){mp[0], mp[1], mp[2], mp[3]});
#pragma unroll
    for (int r = 0; r < 4; ++r) {
      st_nt4(q_cov + idx * 16 + r * 4,
             (v4f){Pn[r * 4 + 0], Pn[r * 4 + 1], Pn[r * 4 + 2], Pn[r * 4 + 3]});
      st_nt4(p_cov + idx * 16 + r * 4,
             (v4f){Pp[r * 4 + 0], Pp[r * 4 + 1], Pp[r * 4 + 2], Pp[r * 4 + 3]});
    }

    // Cholesky of Pn + eps*I, then zhat = mu + u @ L
    const float L00 = sqrtf(Pn[0] + EPS_SAMPLE);
    const float r0 = 1.0f / L00;
    const float L10 = Pn[4] * r0, L20 = Pn[8] * r0, L30 = Pn[12] * r0;
    const float L11 = sqrtf(Pn[5] + EPS_SAMPLE - L10 * L10);
    const float r1 = 1.0f / L11;
    const float L21 = (Pn[9] - L20 * L10) * r1;
    const float L31 = (Pn[13] - L30 * L10) * r1;
    const float L22 = sqrtf(Pn[10] + EPS_SAMPLE - L20 * L20 - L21 * L21);
    const float r2 = 1.0f / L22;
    const float L32 = (Pn[14] - L30 * L20 - L31 * L21) * r2;
    const float L33 = sqrtf(Pn[15] + EPS_SAMPLE - L30 * L30 - L31 * L31 - L32 * L32);

    const float zh0 = mu[0] + u[0] * L00 + u[1] * L10 + u[2] * L20 + u[3] * L30;
    const float zh1 = mu[1] + u[1] * L11 + u[2] * L21 + u[3] * L31;
    const float zh2 = mu[2] + u[2] * L22 + u[3] * L32;
    const float zh3 = mu[3] + u[3] * L33;
    *(v4f*)(zh_ws + idx * 4) = (v4f){zh0, zh1, zh2, zh3};  // re-read: keep RT

    // carry
#pragma unroll
    for (int i = 0; i < 4; ++i) mc[i] = mu[i];
#pragma unroll
    for (int i = 0; i < 16; ++i) P[i] = Pn[i];
  }
}

// ---------------------------------------------------------------------------
// Decoder: 16-position tiles. Layer1 [4->6]: one exact-K WMMA. LDS transpose,
// layer2 [6->10]: two WMMAs (K padded 6->8). Branchless operand/bias builds;
// x_recon stored non-temporal.
// ---------------------------------------------------------------------------
__global__ void __launch_bounds__(256) svae_decoder(
    const float* __restrict__ zh_ws,
    const float* __restrict__ wd1, const float* __restrict__ bd1,
    const float* __restrict__ wd2, const float* __restrict__ bd2,
    float* __restrict__ xr) {
  __shared__ float hbuf[8][256];
  const int lane = threadIdx.x & 31;
  const int wave = threadIdx.x >> 5;
  const int tile = blockIdx.x * 8 + wave;
  const int b  = tile >> 6;
  const int t0 = (tile & 63) << 4;
  const int m  = lane & 15;
  const int hi = lane >> 4;
  const bool mok = (m < 6);
  const int col6 = mok ? m : 0;
  const int colA = (m < OBS) ? m : 0;

  // layer1: A = zhat (K=4 exact), aligned b64 per lane
  v2f a = *(const v2f*)(zh_ws + ((size_t)(t0 + m) * BS + (size_t)b) * 4 + hi * 2);
  const int kz = hi * 2;
  const float wa = wd1[(kz + 0) * 6 + col6];
  const float wb = wd1[(kz + 1) * 6 + col6];
  v2f w;
  w.x = mok ? wa : 0.f;
  w.y = mok ? wb : 0.f;
  v8f c1 = {0.f, 0.f, 0.f, 0.f, 0.f, 0.f, 0.f, 0.f};
  c1 = wmma4(a, w, c1);

  const float b1raw = bd1[col6];            // clamped, unconditional
  const float bias1 = mok ? b1raw : 0.f;
#pragma unroll
  for (int r = 0; r < 8; ++r) {
    float h = c1[r] + bias1;
    hbuf[wave][(r + 8 * hi) * 16 + m] = h > 0.f ? h : 0.f;
  }
  __syncthreads();

  v8f c2 = {0.f, 0.f, 0.f, 0.f, 0.f, 0.f, 0.f, 0.f};
#pragma unroll
  for (int kc = 0; kc < 2; ++kc) {
    const int k0 = kc * 4 + hi * 2;
    // A fragment from LDS: indices always in-bounds, mask values only
    const float a0 = hbuf[wave][m * 16 + k0];
    const float a1 = hbuf[wave][m * 16 + k0 + 1];
    v2f a2;
    a2.x = (k0     < 6) ? a0 : 0.f;
    a2.y = (k0 + 1 < 6) ? a1 : 0.f;
    // B fragment: clamped rows/col, value-masked
    const int rr = (k0 < 6) ? k0 : 0;
    const float w0 = wd2[rr * OBS + colA];
    const float w1v = wd2[(rr + 1) * OBS + colA];
    v2f w2;
    w2.x = (k0     < 6 && m < OBS) ? w0 : 0.f;
    w2.y = (k0 + 1 < 6 && m < OBS) ? w1v : 0.f;
    c2 = wmma4(a2, w2, c2);
  }

  const float bias2 = bd2[colA];            // clamped, unconditional
  if (m < OBS) {
#pragma unroll
    for (int r = 0; r < 8; ++r) {
      const int M = r + 8 * hi;
      __builtin_nontemporal_store(
          c2[r] + bias2, &xr[((size_t)b * TT + (size_t)(t0 + M)) * OBS + m]);
    }
  }
}

// ---------------------------------------------------------------------------
extern "C" void kernel_launch(void* const* d_in, const int* in_sizes, int n_in,
                              void* d_out, int out_size, void* d_ws,
                              size_t ws_size, hipStream_t stream) {
  const float* x    = (const float*)d_in[0];
  const float* ew1  = (const float*)d_in[1];
  const float* eb1  = (const float*)d_in[2];
  const float* ewx  = (const float*)d_in[3];
  const float* ebx  = (const float*)d_in[4];
  const float* ewl  = (const float*)d_in[5];
  const float* ebl  = (const float*)d_in[6];
  const float* dw1  = (const float*)d_in[7];
  const float* db1  = (const float*)d_in[8];
  const float* dw2  = (const float*)d_in[9];
  const float* db2  = (const float*)d_in[10];
  const float* kfA  = (const float*)d_in[11];
  const float* kfb  = (const float*)d_in[12];
  const float* kfQ  = (const float*)d_in[13];
  const int*   seed = (const int*)d_in[14];

  float* out   = (float*)d_out;
  float* zm_ws = (float*)d_ws;          // [T][BS][4]  16 MB
  float* sg_ws = zm_ws + 4 * NPOS;      // [T][BS][4]  16 MB
  float* zh_ws = sg_ws + 4 * NPOS;      // [T][BS][4]  16 MB
  float* u_ws  = zh_ws + 4 * NPOS;      // [T][4]      16 KB
  float* Q_ws  = u_ws + 4 * TT;         // [4][4]

  svae_prep<<<dim3(4), dim3(256), 0, stream>>>(kfQ, seed, u_ws, Q_ws);
  svae_encoder<<<dim3(8192), dim3(256), 0, stream>>>(x, ew1, eb1, ewx, ebx, ewl,
                                                     ebl, zm_ws, sg_ws);
  svae_scan<<<dim3(8), dim3(128), 0, stream>>>(zm_ws, sg_ws, u_ws, Q_ws, kfA,
                                               kfb, out, zh_ws);
  svae_decoder<<<dim3(8192), dim3(256), 0, stream>>>(zh_ws, dw1, db1, dw2, db2,
                                                     out);
}